// SparseGRUBrain_72962904424823
// MI455X (gfx1250) — compile-verified
//
#include <hip/hip_runtime.h>

// ---------------------------------------------------------------------------
// SparseGRUBrain for MI455X (gfx1250, wave32)
//   Roofline: ~270 MB of traffic ~ 12us at 23.3 TB/s; FLOPs negligible.
//   Key design points:
//   * COO scatter-add (215M f32 atomic RMWs) replaced by on-device CSR build
//     (~2.3M u32 atomics) + register-accumulating gather: no RMW serialization,
//     no 54MB zero-fill pass.
//   * Per-neuron 8x8x8 GEMMs run on V_WMMA_F32_16X16X4_F32, two neurons packed
//     block-diagonally per 16x16 wave tile; only A is block-masked (off-diagonal
//     garbage lands in lanes 8-23, never read).
//   * Scatter accumulators in (N,H,B) layout: gather stores and GRU loads are
//     contiguous float4s.
// ---------------------------------------------------------------------------

static constexpr int N = 70000;
static constexpr int H = 8;
static constexpr int E = 1120000;
static constexpr int B = 8;
static constexpr size_t NHB = (size_t)N * H * B;   // 4,480,000 floats per array

typedef __attribute__((ext_vector_type(2))) float v2f;
typedef __attribute__((ext_vector_type(8))) float v8f;

__device__ __forceinline__ float fast_sigmoid(float x) {
  return __builtin_amdgcn_rcpf(1.0f + __expf(-x));   // v_exp_f32 + v_rcp_f32
}
__device__ __forceinline__ float fast_tanh(float x) {
  // 1 - 2/(1+exp(2x)); exp->inf => 1, exp->0 => -1 (correct saturation)
  return 1.0f - 2.0f * __builtin_amdgcn_rcpf(1.0f + __expf(2.0f * x));
}

// ---------------------------------------------------------------------------
// CSR build: zero counts -> count degrees -> exclusive scan -> stable-ish fill
// ---------------------------------------------------------------------------
__global__ void zero_u32_kernel(unsigned* __restrict__ p, int n) {
  int i = blockIdx.x * blockDim.x + threadIdx.x;
  if (i < n) p[i] = 0u;
}

__global__ __launch_bounds__(256) void count_kernel(
    const int* __restrict__ tgt, unsigned* __restrict__ cnt) {
  int e = blockIdx.x * blockDim.x + threadIdx.x;
  if (e < E) atomicAdd(&cnt[tgt[e]], 1u);
}

// Single-workgroup scan over N=70000 degrees: each of 1024 threads sums a
// contiguous chunk, Hillis-Steele scan of the 1024 partials in LDS, then each
// thread emits the running exclusive prefix over its chunk.
__global__ __launch_bounds__(1024) void scan_kernel(
    const unsigned* __restrict__ cnt, unsigned* __restrict__ off,
    unsigned* __restrict__ cursor) {
  __shared__ unsigned ssum[1024];
  const int t = threadIdx.x;
  const int chunk = (N + 1023) / 1024;               // 69
  const int lo = t * chunk;
  const int hi = (lo + chunk < N) ? (lo + chunk) : N;
  unsigned s = 0;
  for (int i = lo; i < hi; ++i) s += cnt[i];
  ssum[t] = s;
  __syncthreads();
  for (int d = 1; d < 1024; d <<= 1) {
    unsigned o = (t >= d) ? ssum[t - d] : 0u;
    __syncthreads();
    ssum[t] += o;
    __syncthreads();
  }
  unsigned run = ssum[t] - s;                        // exclusive prefix
  for (int i = lo; i < hi; ++i) {
    off[i] = run;
    cursor[i] = run;
    run += cnt[i];
  }
  if (t == 1023) off[N] = run;                       // == E
}

__global__ __launch_bounds__(256) void fill_kernel(
    const int* __restrict__ tgt, unsigned* __restrict__ cursor,
    int* __restrict__ eid) {
  int e = blockIdx.x * blockDim.x + threadIdx.x;
  if (e < E) {
    unsigned pos = atomicAdd(&cursor[tgt[e]], 1u);
    eid[pos] = e;
  }
}

// ---------------------------------------------------------------------------
// CSR gather: inp_*[n,h,b] = sum_{e in seg(n)} W_*[e,h] * cal[b, src[e]]
// 8 threads per neuron (one per h); all accumulation in VGPRs, no atomics.
// W row reads coalesce 32B across the 8-lane group; cal loads are group-
// uniform (hardware-broadcast, L2-resident). Stores: contiguous float4 pairs.
// ---------------------------------------------------------------------------
__global__ __launch_bounds__(256) void gather_kernel(
    const float* __restrict__ cal, const int* __restrict__ src,
    const unsigned* __restrict__ off, const int* __restrict__ eid,
    const float* __restrict__ Wz, const float* __restrict__ Wr,
    const float* __restrict__ Wh,
    float* __restrict__ inp_z, float* __restrict__ inp_r,
    float* __restrict__ inp_h) {
  const int tid = blockIdx.x * blockDim.x + threadIdx.x;
  const int n = tid >> 3;
  const int h = tid & 7;
  if (n >= N) return;

  const unsigned beg = off[n];
  const unsigned end = off[n + 1];

  float az[8], ar[8], ah[8];
#pragma unroll
  for (int b = 0; b < B; ++b) { az[b] = 0.f; ar[b] = 0.f; ah[b] = 0.f; }

  for (unsigned k = beg; k < end; ++k) {
    const int e = eid[k];
    const int s = src[e];
    const float wz = Wz[(size_t)e * H + h];
    const float wr = Wr[(size_t)e * H + h];
    const float wh = Wh[(size_t)e * H + h];
#pragma unroll
    for (int b = 0; b < B; ++b) {
      const float c = cal[(size_t)b * N + s];
      az[b] = fmaf(wz, c, az[b]);
      ar[b] = fmaf(wr, c, ar[b]);
      ah[b] = fmaf(wh, c, ah[b]);
    }
  }

  const size_t o = ((size_t)n * H + h) * B;
  *(float4*)(inp_z + o)     = make_float4(az[0], az[1], az[2], az[3]);
  *(float4*)(inp_z + o + 4) = make_float4(az[4], az[5], az[6], az[7]);
  *(float4*)(inp_r + o)     = make_float4(ar[0], ar[1], ar[2], ar[3]);
  *(float4*)(inp_r + o + 4) = make_float4(ar[4], ar[5], ar[6], ar[7]);
  *(float4*)(inp_h + o)     = make_float4(ah[0], ah[1], ah[2], ah[3]);
  *(float4*)(inp_h + o + 4) = make_float4(ah[4], ah[5], ah[6], ah[7]);
}

// ---------------------------------------------------------------------------
// Fused GRU update. One wave32 handles a neuron pair (n0, n1).
//
// 16x16 WMMA tile, K=16 via 4 chained K=4 f32 WMMAs per U matrix:
//   A rows 0-7  = hidden[b, n0, k]   (k<8;  zero-masked elsewhere)
//   A rows 8-15 = hidden[b, n1, k-8] (k>=8; zero-masked elsewhere)
//   B cols j    = U[n0|n1][k&7][j&7]  (UNMASKED: off-diagonal garbage only
//                                      reaches lanes 8-23, never read)
// A-layout (32-bit A 16x4): lane L holds M=L&15; VGPR v holds
//   K = 4*chunk + (L<16 ? 0 : 2) + v  => each lane touches only
//   k&7 in {klo, klo+1, klo+4, klo+5} (4 values).
// C-layout: lane L holds column N=L&15; acc[v] is row M=v+(L<16?0:8)
//   => useful lanes 0-7 (n0) and 24-31 (n1), batch across the 8 acc VGPRs.
// ---------------------------------------------------------------------------
__global__ __launch_bounds__(256) void gru_wmma_kernel(
    const float* __restrict__ hidden,
    const float* __restrict__ Uz, const float* __restrict__ Ur,
    const float* __restrict__ Uh,
    const float* __restrict__ bz, const float* __restrict__ br,
    const float* __restrict__ bh,
    const float* __restrict__ proj,
    const float* __restrict__ inp_z, const float* __restrict__ inp_r,
    const float* __restrict__ inp_h,
    float* __restrict__ out_cal, float* __restrict__ out_hid) {
  __shared__ float s_rh[8][2][8][8];  // [wave][nsel][b][h]  (r * hidden)

  const int wv   = threadIdx.x >> 5;
  const int lane = threadIdx.x & 31;
  const int pair = blockIdx.x * 8 + wv;  // grid sized exactly: pair < N/2
  const int n0   = pair * 2;

  // ---- A-layout lane constants ----
  const int m      = lane & 15;
  const int b_a    = m & 7;
  const int nsel_a = m >> 3;
  const int n_a    = n0 + nsel_a;
  const int klo    = (lane < 16) ? 0 : 2;
  const bool pA0   = (nsel_a == 0);

  float a4[4];  // hidden[b_a, n_a, {klo, klo+1, klo+4, klo+5}]
  {
    const float* hp = hidden + ((size_t)b_a * N + n_a) * H + klo;
    float2 h0 = *(const float2*)(hp);
    float2 h1 = *(const float2*)(hp + 4);
    a4[0] = h0.x; a4[1] = h0.y; a4[2] = h1.x; a4[3] = h1.y;
  }

  // ---- B-layout lane constants: lane owns one output column ----
  const int jcol   = lane & 15;
  const int nsel_b = jcol >> 3;
  const int col    = jcol & 7;
  const int n_b    = n0 + nsel_b;
  float uz4[4], ur4[4], uh4[4];  // U[n_b, {klo,klo+1,klo+4,klo+5}, col]
  {
    const float* pz = Uz + (size_t)n_b * (H * H) + col;
    const float* pr = Ur + (size_t)n_b * (H * H) + col;
    const float* ph = Uh + (size_t)n_b * (H * H) + col;
#pragma unroll
    for (int j = 0; j < 4; ++j) {
      const int hh = klo + (j & 1) + (j >> 1) * 4;
      uz4[j] = pz[hh * H];
      ur4[j] = pr[hh * H];
      uh4[j] = ph[hh * H];
    }
  }

  // ---- rec_z / rec_r : chained f32 WMMAs (uniform control flow) ----
  v8f accz = {0.f,0.f,0.f,0.f,0.f,0.f,0.f,0.f};
  v8f accr = {0.f,0.f,0.f,0.f,0.f,0.f,0.f,0.f};
#pragma unroll
  for (int c = 0; c < 4; ++c) {
    const bool hi = (c >= 2);
    const int  ix = (c & 1) * 2;
    v2f a, bzv, brv;
    a[0] = (pA0 != hi) ? a4[ix + 0] : 0.0f;   // mask A only
    a[1] = (pA0 != hi) ? a4[ix + 1] : 0.0f;
    bzv[0] = uz4[ix + 0]; bzv[1] = uz4[ix + 1];
    brv[0] = ur4[ix + 0]; brv[1] = ur4[ix + 1];
    accz = __builtin_amdgcn_wmma_f32_16x16x4_f32(false, a, false, bzv,
                                                 (short)0, accz, false, false);
    accr = __builtin_amdgcn_wmma_f32_16x16x4_f32(false, a, false, brv,
                                                 (short)0, accr, false, false);
  }

  // ---- C-layout elementwise: lane owns fixed (n_c, i_c), batch in acc[v] ----
  const bool active = (lane < 8) || (lane >= 24);
  const int nsel_c  = (lane >= 24) ? 1 : 0;
  const int n_c     = n0 + nsel_c;
  const int i_c     = lane & 7;
  const size_t ni   = (size_t)n_c * H + i_c;

  float izv[8], irv[8], ihv[8];  // (N,H,B): 8 batch values 32B contiguous
  {
    const size_t o = ((size_t)n_c * H + i_c) * B;
    const float4* qz = (const float4*)(inp_z + o);
    const float4* qr = (const float4*)(inp_r + o);
    const float4* qh = (const float4*)(inp_h + o);
    float4 t0 = qz[0], t1 = qz[1];
    izv[0]=t0.x; izv[1]=t0.y; izv[2]=t0.z; izv[3]=t0.w;
    izv[4]=t1.x; izv[5]=t1.y; izv[6]=t1.z; izv[7]=t1.w;
    t0 = qr[0]; t1 = qr[1];
    irv[0]=t0.x; irv[1]=t0.y; irv[2]=t0.z; irv[3]=t0.w;
    irv[4]=t1.x; irv[5]=t1.y; irv[6]=t1.z; irv[7]=t1.w;
    t0 = qh[0]; t1 = qh[1];
    ihv[0]=t0.x; ihv[1]=t0.y; ihv[2]=t0.z; ihv[3]=t0.w;
    ihv[4]=t1.x; ihv[5]=t1.y; ihv[6]=t1.z; ihv[7]=t1.w;
  }

  const float bzs = bz[ni], brs = br[ni], bhs = bh[ni];
  float zv[8], hidc[8];
#pragma unroll
  for (int b = 0; b < B; ++b) {
    hidc[b] = hidden[(size_t)b * (size_t)N * H + ni];
    zv[b]   = fast_sigmoid(izv[b] + accz[b] + bzs);
    const float r = fast_sigmoid(irv[b] + accr[b] + brs);
    if (active) s_rh[wv][nsel_c][b][i_c] = r * hidc[b];
  }
  __syncthreads();  // uniform: full blocks, no divergent returns

  // ---- rec_h WMMA with A' = r * hidden (transposed back via LDS) ----
  float rh4[4];
  {
    const float* sp = &s_rh[wv][nsel_a][b_a][0] + klo;
    float2 r0 = *(const float2*)(sp);
    float2 r1 = *(const float2*)(sp + 4);
    rh4[0] = r0.x; rh4[1] = r0.y; rh4[2] = r1.x; rh4[3] = r1.y;
  }

  v8f acch = {0.f,0.f,0.f,0.f,0.f,0.f,0.f,0.f};
#pragma unroll
  for (int c = 0; c < 4; ++c) {
    const bool hi = (c >= 2);
    const int  ix = (c & 1) * 2;
    v2f a, bhv;
    a[0] = (pA0 != hi) ? rh4[ix + 0] : 0.0f;
    a[1] = (pA0 != hi) ? rh4[ix + 1] : 0.0f;
    bhv[0] = uh4[ix + 0]; bhv[1] = uh4[ix + 1];
    acch = __builtin_amdgcn_wmma_f32_16x16x4_f32(false, a, false, bhv,
                                                 (short)0, acch, false, false);
  }

  // ---- h_tilde, blend, store hidden_new; projection partials ----
  const float pv = proj[i_c];
  float dot[8];
#pragma unroll
  for (int b = 0; b < B; ++b) {
    const float ht = fast_tanh(ihv[b] + acch[b] + bhs);
    const float hn = (1.0f - zv[b]) * hidc[b] + zv[b] * ht;
    if (active) out_hid[(size_t)b * (size_t)N * H + ni] = hn;
    dot[b] = hn * pv;
  }

  // reduce over i across each aligned 8-lane group (lanes 0-7 / 24-31)
#pragma unroll
  for (int b = 0; b < B; ++b) {
    float s = dot[b];
    s += __shfl_xor(s, 1, 8);
    s += __shfl_xor(s, 2, 8);
    s += __shfl_xor(s, 4, 8);
    dot[b] = s;
  }
  if (active && i_c == 0) {
#pragma unroll
    for (int b = 0; b < B; ++b) {
      const float s = dot[b];
      out_cal[(size_t)b * N + n_c] = s > 0.0f ? s : 0.0f;
    }
  }
}

// ---------------------------------------------------------------------------
extern "C" void kernel_launch(void* const* d_in, const int* in_sizes, int n_in,
                              void* d_out, int out_size, void* d_ws,
                              size_t ws_size, hipStream_t stream) {
  const float* calcium = (const float*)d_in[0];
  const float* hidden  = (const float*)d_in[1];
  const int*   src     = (const int*)d_in[2];
  const int*   tgt     = (const int*)d_in[3];
  const float* Wz      = (const float*)d_in[4];
  const float* Wr      = (const float*)d_in[5];
  const float* Wh      = (const float*)d_in[6];
  const float* Uz      = (const float*)d_in[7];
  const float* Ur      = (const float*)d_in[8];
  const float* Uh      = (const float*)d_in[9];
  const float* bz      = (const float*)d_in[10];
  const float* br      = (const float*)d_in[11];
  const float* bh      = (const float*)d_in[12];
  const float* proj    = (const float*)d_in[13];

  float* out_cal = (float*)d_out;                 // (B, N)
  float* out_hid = out_cal + (size_t)B * N;       // (B, N, H)

  // workspace layout
  float*    inp_z  = (float*)d_ws;                // 3 x (N, H, B) f32
  float*    inp_r  = inp_z + NHB;
  float*    inp_h  = inp_r + NHB;
  unsigned* cnt    = (unsigned*)(inp_h + NHB);    // N
  unsigned* off    = cnt + N;                     // N + 1
  unsigned* cursor = off + (N + 1);               // N
  int*      eid    = (int*)(cursor + N);          // E

  // 1) CSR build: counts -> scan -> fill (u32 atomics only, ~2.3M total)
  zero_u32_kernel<<<(N + 255) / 256, 256, 0, stream>>>(cnt, N);
  count_kernel<<<(E + 255) / 256, 256, 0, stream>>>(tgt, cnt);
  scan_kernel<<<1, 1024, 0, stream>>>(cnt, off, cursor);
  fill_kernel<<<(E + 255) / 256, 256, 0, stream>>>(tgt, cursor, eid);

  // 2) register-accumulating gather (no f32 atomics, no zero-fill pass)
  gather_kernel<<<((N * 8) + 255) / 256, 256, 0, stream>>>(
      calcium, src, off, eid, Wz, Wr, Wh, inp_z, inp_r, inp_h);

  // 3) fused WMMA GRU: N/2 = 35,000 pairs, 8 waves/block -> 4375 blocks exact
  gru_wmma_kernel<<<(N / 2) / 8, 256, 0, stream>>>(
      hidden, Uz, Ur, Uh, bz, br, bh, proj, inp_z, inp_r, inp_h,
      out_cal, out_hid);
}